// PolynomialManyBodyBlock_49091476194032
// MI455X (gfx1250) — compile-verified
//
#include <hip/hip_runtime.h>

typedef float v2f __attribute__((ext_vector_type(2)));
typedef float v4f __attribute__((ext_vector_type(4)));
typedef float v8f __attribute__((ext_vector_type(8)));

// o3.Linear with identical irreps in/out: per-l channel mixing
//   y[(n,i), v] = (1/sqrt(128)) * sum_u x[(n,i), u] * w_l[u, v]
// Template on l (d = 2l+1 compile-time) and NS (samples per block), chosen so
// every block has a comparable row-tile height and >=4 independent WMMA
// accumulator chains per wave.
template <int L, int NS>
__global__ __launch_bounds__(256) void o3lin_kernel(
    const float* __restrict__ x, const float* __restrict__ w,
    float* __restrict__ y, int N)
{
    constexpr int D      = 2 * L + 1;
    constexpr int OFF    = (L == 0) ? 0 : (L == 1) ? 128 : (L == 2) ? 512 : 1152;
    constexpr int DIM    = 2048;           // total feature width
    constexpr int BM     = NS * D;         // row-tile height (multiple of 16)
    constexpr int RT     = BM / 16;        // 16-row tiles per column tile (per wave)
    constexpr int LDSROW = 132;            // padded row stride (floats) -> conflict-free b64 A loads
    constexpr int CHUNK  = 128 * D;        // contiguous floats per sample for this l (mult of 4)
    constexpr int TOTAL  = NS * CHUNK;
    constexpr int TOTAL4 = TOTAL / 4;

    static_assert(BM % 16 == 0, "row tile must be multiple of 16");
    static_assert(BM * LDSROW * 4 <= 64 * 1024, "static LDS budget");

    __shared__ __align__(16) float lds[BM * LDSROW];

    const int tid  = threadIdx.x;
    const int lane = tid & 31;
    const int wave = tid >> 5;             // 8 waves -> 8 column tiles of 16
    const int lv   = lane & 15;
    const int hi   = lane >> 4;            // 0: lanes 0-15 (K=0,1), 1: lanes 16-31 (K=2,3)
    const int n0   = blockIdx.x * NS;

    // ---- stage X tile: b128 coalesced global reads, transposed scatter to [row=(s*D+i)][u] ----
    for (int t4 = tid; t4 < TOTAL4; t4 += 256) {
        const int t = t4 * 4;
        const int s = t / CHUNK;
        const int r = t - s * CHUNK;       // r = u*D + i within the sample chunk
        v4f val = {0.0f, 0.0f, 0.0f, 0.0f};
        if (n0 + s < N)
            val = *(const v4f*)&x[(size_t)(n0 + s) * DIM + OFF + r];
#pragma unroll
        for (int e = 0; e < 4; ++e) {
            const int re = r + e;
            const int u  = re / D;
            const int i  = re - u * D;
            lds[(s * D + i) * LDSROW + u] = val[e];
        }
    }

    // ---- preload this wave's 32 B-fragments of W (columns wave*16 .. +15) into VGPRs ----
    // B layout (16x16x4 f32): vgpr0 = B[k0, n], vgpr1 = B[k0+1, n];
    // lanes 0-15: k0 = 4*kk, lanes 16-31: k0 = 4*kk + 2; n = lane%16.
    const int v = wave * 16 + lv;
    v2f breg[32];
#pragma unroll
    for (int kk = 0; kk < 32; ++kk) {
        const int k0 = kk * 4 + hi * 2;
        breg[kk].x = w[(size_t)k0 * 128 + v];
        breg[kk].y = w[(size_t)(k0 + 1) * 128 + v];
    }

    __syncthreads();

    // ---- K=128 accumulation via 32 x V_WMMA_F32_16X16X4_F32 per 16x16 output tile;
    //      RT independent accumulator chains per wave hide WMMA latency ----
    v8f acc[RT];
#pragma unroll
    for (int rt = 0; rt < RT; ++rt)
#pragma unroll
        for (int j = 0; j < 8; ++j)
            acc[rt][j] = 0.0f;

#pragma unroll
    for (int kk = 0; kk < 32; ++kk) {
        const v2f b = breg[kk];
#pragma unroll
        for (int rt = 0; rt < RT; ++rt) {
            // A layout: vgpr0 = A[m, k0], vgpr1 = A[m, k0+1]; m = lane%16,
            // lanes 0-15: k0 = 4*kk, lanes 16-31: k0 = 4*kk + 2.  8B-aligned ds_load_b64.
            const v2f a = *(const v2f*)&lds[(rt * 16 + lv) * LDSROW + kk * 4 + hi * 2];
            acc[rt] = __builtin_amdgcn_wmma_f32_16x16x4_f32(
                /*neg_a=*/false, a, /*neg_b=*/false, b,
                /*c_mod=*/(short)0, acc[rt], /*reuse_a=*/false, /*reuse_b=*/false);
        }
    }

    __syncthreads();

    // ---- scale (after reduction, matching reference) and restage to e3nn output layout ----
    constexpr float INV_SQRT_MUL = 0.08838834764831843f; // 1/sqrt(128)
    // C/D layout: VGPR j holds M = j (lanes 0-15) / M = j+8 (lanes 16-31), N = lane%16.
#pragma unroll
    for (int rt = 0; rt < RT; ++rt) {
#pragma unroll
        for (int j = 0; j < 8; ++j) {
            const int m   = j + hi * 8;
            const int row = rt * 16 + m;   // = s*D + i
            const int s   = row / D;
            const int i   = row - s * D;
            lds[s * CHUNK + v * D + i] = acc[rt][j] * INV_SQRT_MUL;
        }
    }

    __syncthreads();

    // ---- fully coalesced b128 store of the NS-sample output tile ----
    for (int t4 = tid; t4 < TOTAL4; t4 += 256) {
        const int t = t4 * 4;
        const int s = t / CHUNK;
        const int r = t - s * CHUNK;
        if (n0 + s < N) {
            const v4f val = *(const v4f*)&lds[t];
            *(v4f*)&y[(size_t)(n0 + s) * DIM + OFF + r] = val;
        }
    }
}

extern "C" void kernel_launch(void* const* d_in, const int* in_sizes, int n_in,
                              void* d_out, int out_size, void* d_ws, size_t ws_size,
                              hipStream_t stream) {
    // setup_inputs order: node_feats, sc, node_attrs, w0, w1, w2, w3, use_direct
    const float* x  = (const float*)d_in[0];
    const float* w0 = (const float*)d_in[3];
    const float* w1 = (const float*)d_in[4];
    const float* w2 = (const float*)d_in[5];
    const float* w3 = (const float*)d_in[6];
    float* y = (float*)d_out;

    const int N = in_sizes[0] / 2048;

    // NS per l: keep BM = NS*(2l+1) in [64,112] rows, static LDS <= 64KB.
    o3lin_kernel<0, 64><<<(N + 63) / 64, 256, 0, stream>>>(x, w0, y, N);
    o3lin_kernel<1, 32><<<(N + 31) / 32, 256, 0, stream>>>(x, w1, y, N);
    o3lin_kernel<2, 16><<<(N + 15) / 16, 256, 0, stream>>>(x, w2, y, N);
    o3lin_kernel<3, 16><<<(N + 15) / 16, 256, 0, stream>>>(x, w3, y, N);
}